// TPTransMatrix_57810259804235
// MI455X (gfx1250) — compile-verified
//
#include <hip/hip_runtime.h>
#include <stdint.h>

// Grouped block-diagonal GEMM for MI455X (gfx1250, wave32, WMMA).
// Hybrid staging: A tile via CDNA5 async global->LDS (ASYNCcnt, double-buffered),
// B tile via pipelined global_load_b128 -> K-transposed LDS (mov-free b64 fragments).
// x:      [16384, 4096] fp32 (4*4096 tokens)
// blocks: [8, 512, 512] fp32
// out[t, g*512+n] = sum_k x[t, g*512+k] * blocks[g][k][n]

typedef __attribute__((ext_vector_type(2))) float v2f;
typedef __attribute__((ext_vector_type(4))) float v4f;
typedef __attribute__((ext_vector_type(8))) float v8f;

#define HIDDEN 4096
#define NBLK   8
#define BLK    512
#define TOKENS 16384

#define BM 128
#define BN 64
#define BK 32
#define NC (BLK / BK)   // 16 K-chunks

#define LDA 36          // A rows (floats): 144 B -> every 16B async write aligned, b64 frag reads aligned
#define LDB 34          // transposed B rows (floats): 136 B -> b64 frag reads aligned

// ---- CDNA5 async global->LDS copy (ASYNCcnt-tracked), GV addressing form ----
__device__ __forceinline__ void async_copy_b128(unsigned lds_off, const float* gptr) {
    asm volatile("global_load_async_to_lds_b128 %0, %1, off"
                 :: "v"(lds_off), "v"((unsigned long long)(uintptr_t)gptr)
                 : "memory");
}
__device__ __forceinline__ void async_wait_0() {
    asm volatile("s_wait_asynccnt 0x0" ::: "memory");
}

__global__ __launch_bounds__(256)
void TPTransMatrix_grouped_wmma_f32(const float* __restrict__ x,
                                    const float* __restrict__ blocks,
                                    float* __restrict__ out)
{
    __shared__ __align__(16) float As[2][BM][LDA];   // As[buf][m][k]   (async target)
    __shared__ float Bs[2][BN][LDB];                 // Bs[buf][n][k]   (K-transposed)

    const int g     = blockIdx.z;        // block-diagonal group 0..7
    const int mtile = blockIdx.y;        // 0..127
    const int ntile = blockIdx.x;        // 0..7

    const int tid  = threadIdx.x;
    const int lane = tid & 31;
    const int wave = tid >> 5;           // 0..7
    const int lhi  = lane >> 4;          // 0/1  : K-pair (A/B) or M+8 (C/D) selector
    const int llo  = lane & 15;          // M (A) / N (B,C,D) index within tile

    const int    m_base = mtile * BM;
    const int    nb     = ntile * BN;                       // n offset inside the 512 block
    const float* xg     = x + (size_t)m_base * HIDDEN + (size_t)g * BLK;
    const float* Bg     = blocks + (size_t)g * BLK * BLK;

    // LDS byte offset of As (low 32 bits of generic pointer == ds address)
    const unsigned ldsA = (unsigned)(unsigned long long)(uintptr_t)&As[0][0][0];
    const unsigned ABUF = BM * LDA * 4;                     // bytes per A buffer

    // ---- A: 4 async b128 per thread per chunk (128x32 floats) ----
    auto issue_A = [&](int c, int buf) {
        const int k0 = c * BK;
        #pragma unroll
        for (int p = 0; p < 4; ++p) {
            int lin = p * 256 + tid;                        // 0..1023 float4 slots
            int row = lin >> 3;                             // 0..127
            int c4  = (lin & 7) << 2;                       // 0,4,...,28
            async_copy_b128(ldsA + (unsigned)buf * ABUF + (unsigned)(row * LDA + c4) * 4,
                            xg + (size_t)row * HIDDEN + k0 + c4);
        }
    };

    // ---- B: 2 b128 per thread per chunk (32x64 floats), VGPR staged ----
    const int krow0 = tid >> 4;                             // 0..15 (p=0), +16 for p=1
    const int bc4   = (tid & 15) << 2;                      // 0,4,...,60
    auto load_B = [&](int c, int p) -> v4f {
        return *(const v4f*)(Bg + (size_t)(c * BK + p * 16 + krow0) * BLK + nb + bc4);
    };
    auto store_B = [&](v4f v, int p, int buf) {             // transpose into Bs[n][k]
        int krow = p * 16 + krow0;
        Bs[buf][bc4 + 0][krow] = v.x;
        Bs[buf][bc4 + 1][krow] = v.y;
        Bs[buf][bc4 + 2][krow] = v.z;
        Bs[buf][bc4 + 3][krow] = v.w;
    };

    v8f acc[4];
    {
        v8f zero = {};
        #pragma unroll
        for (int t = 0; t < 4; ++t) acc[t] = zero;
    }

    const int wm = wave * 16;            // this wave's M strip inside the BM tile

    // ---- prologue: stage chunk 0 ----
    issue_A(0, 0);
    {
        v4f b0 = load_B(0, 0);
        v4f b1 = load_B(0, 1);
        store_B(b0, 0, 0);
        store_B(b1, 1, 0);
    }
    async_wait_0();
    __syncthreads();

    for (int c = 0; c < NC; ++c) {
        const int  buf  = c & 1;
        const bool more = (c + 1 < NC);

        v4f nb0, nb1;
        if (more) {
            issue_A(c + 1, buf ^ 1);     // async A copies fly during compute
            nb0 = load_B(c + 1, 0);      // B prefetch into VGPRs
            nb1 = load_B(c + 1, 1);
        }

        // ---- 8 K-steps of 4, each A fragment reused by 4 WMMAs ----
        #pragma unroll
        for (int kk = 0; kk < BK; kk += 4) {
            // A 16x4 fragment: lanes 0-15 hold K=kk,kk+1; lanes 16-31 hold K=kk+2,kk+3
            v2f a = *(const v2f*)&As[buf][wm + llo][kk + 2 * lhi];
            #pragma unroll
            for (int t = 0; t < 4; ++t) {
                // B 4x16 fragment: contiguous (K,K+1) pair in transposed Bs -> single b64
                v2f b = *(const v2f*)&Bs[buf][t * 16 + llo][kk + 2 * lhi];
                acc[t] = __builtin_amdgcn_wmma_f32_16x16x4_f32(
                    /*neg_a=*/false, a, /*neg_b=*/false, b,
                    /*c_mod=*/(short)0, acc[t],
                    /*reuse_a=*/false, /*reuse_b=*/false);
            }
        }

        if (more) {
            store_B(nb0, 0, buf ^ 1);    // other buffer: its readers finished last barrier
            store_B(nb1, 1, buf ^ 1);
            async_wait_0();              // A(c+1) landed (overlap already happened)
        }
        __syncthreads();                 // chunk c+1 fully visible; buffers safe to reuse
    }

    // ---- store: C/D layout -> VGPR r : M=r (lanes 0-15) / M=r+8 (lanes 16-31), N=llo
    float* op = out + (size_t)(m_base + wm) * HIDDEN + (size_t)g * BLK + nb;
    #pragma unroll
    for (int t = 0; t < 4; ++t) {
        #pragma unroll
        for (int r = 0; r < 8; ++r) {
            op[(size_t)(r + 8 * lhi) * HIDDEN + t * 16 + llo] = acc[t][r];
        }
    }
}

extern "C" void kernel_launch(void* const* d_in, const int* in_sizes, int n_in,
                              void* d_out, int out_size, void* d_ws, size_t ws_size,
                              hipStream_t stream) {
    const float* x      = (const float*)d_in[0];   // [4,4096,4096]
    const float* blocks = (const float*)d_in[1];   // [8,512,512]
    float*       out    = (float*)d_out;           // [4,4096,4096]

    dim3 grid(BLK / BN, TOKENS / BM, NBLK);        // (8, 128, 8)
    TPTransMatrix_grouped_wmma_f32<<<grid, 256, 0, stream>>>(x, blocks, out);
}